// YOLOv3Loss_66563403153817
// MI455X (gfx1250) — compile-verified
//
#include <hip/hip_runtime.h>
#include <math.h>

typedef float v2f __attribute__((ext_vector_type(2)));
typedef float v8f __attribute__((ext_vector_type(8)));

#define NCLS 20
#define IMG 480.0f
#define TILE_CELLS 256
#define TILE_FLOATS (TILE_CELLS * 25)        // 6400 floats = 25600 B per block
#define TILE_CHUNKS (TILE_FLOATS / 4)        // 1600 x b128

// Exact mirror of the reference's clamped-log BCE.
__device__ __forceinline__ float bce(float p, float t) {
    float lp = (p > 0.0f) ? fmaxf(logf(p), -100.0f) : -100.0f;
    float lq = (p < 1.0f) ? fmaxf(logf(1.0f - p), -100.0f) : -100.0f;
    return -(t * lp + (1.0f - t) * lq);
}

// ws[0..partialInts) = 0.0f (block partials), rest = -1 (winner indices)
__global__ void yolo_init(int* __restrict__ ws, int partialInts, int totalInts) {
    int i = blockIdx.x * blockDim.x + threadIdx.x;
    if (i < totalInts) ws[i] = (i < partialInts) ? 0 : -1;
}

// Phase A: scatter ground-truth boxes to grid cells; "largest m wins" via atomicMax,
// emulating last-write-wins of the reference scatter.
__global__ void yolo_build(const float* __restrict__ gts,
                           int* __restrict__ w1, int* __restrict__ w2, int* __restrict__ w3) {
    int tid = blockIdx.x * blockDim.x + threadIdx.x;
    if (tid >= 64 * 300) return;
    int b = tid / 300, m = tid % 300;
    const float* g = gts + (size_t)tid * 25;
    if (!(g[4] > 0.5f)) return;
    float mx = g[0], my = g[1];
    float w = g[2] * IMG, h = g[3] * IMG;
    const float anc[3][3][2] = {
        {{10.f,13.f},{16.f,30.f},{33.f,23.f}},
        {{30.f,61.f},{62.f,45.f},{59.f,119.f}},
        {{116.f,90.f},{156.f,198.f},{373.f,326.f}}};
    const int Gs[3] = {60, 30, 15};
    int* Ws[3] = {w1, w2, w3};
    for (int s = 0; s < 3; ++s) {
        int G = Gs[s];
        float bestIou = -1.0f; int best = 0;
        for (int a = 0; a < 3; ++a) {
            float aw = anc[s][a][0], ah = anc[s][a][1];
            float inter = fminf(w, aw) * fminf(h, ah);
            float uni = w * h + aw * ah - inter;
            float iou = inter / (uni + 1e-16f);
            if (iou > bestIou) { bestIou = iou; best = a; }   // first-max == jnp.argmax
        }
        int gx = (int)floorf(mx * (float)G);
        int gy = (int)floorf(my * (float)G);
        gx = gx < 0 ? 0 : (gx > G - 1 ? G - 1 : gx);
        gy = gy < 0 ? 0 : (gy > G - 1 ? G - 1 : gy);
        int idx = ((b * 3 + best) * G + gy) * G + gx;
        atomicMax(&Ws[s][idx], m);
    }
}

// Phase B: each block stages its 256 contiguous cells (6400 floats) into LDS via
// coalesced b128 async copies (ASYNCcnt path), then computes per-cell losses and
// a deterministic block-level tree reduction into per-block partials.
__global__ void __launch_bounds__(256)
yolo_scale_loss(const float* __restrict__ pred,
                const float* __restrict__ gts,
                const int* __restrict__ winner,
                float* __restrict__ partials,
                int G, int numCells, int blockOffset,
                float aw0, float aw1, float aw2,
                float ah0, float ah1, float ah2) {
    __shared__ float tile[TILE_FLOATS];
    __shared__ float s0[256], s1[256], s2[256], s3[256];

    int tid = threadIdx.x;
    int blockFloatBase = blockIdx.x * TILE_FLOATS;

    // Valid floats for this block (scale-3 tail); always a multiple of 4.
    int validFloats = numCells * 25 - blockFloatBase;
    if (validFloats > TILE_FLOATS) validFloats = TILE_FLOATS;
    int validChunks = validFloats >> 2;

    // Async global -> LDS staging, 16B per lane per issue, fully coalesced.
    for (int chunk = tid; chunk < validChunks; chunk += 256) {
        const float* gptr = pred + (size_t)blockFloatBase + (size_t)chunk * 4;
        unsigned ldsOff = (unsigned)(size_t)(&tile[0]) + (unsigned)chunk * 16u;
        asm volatile("global_load_async_to_lds_b128 %0, %1, off"
                     :: "v"(ldsOff), "v"(gptr) : "memory");
    }
    asm volatile("s_wait_asynccnt 0x0" ::: "memory");
    __syncthreads();

    int cell = blockIdx.x * TILE_CELLS + tid;
    float cAcc = 0.0f, oAcc = 0.0f, nAcc = 0.0f, clAcc = 0.0f;
    if (cell < numCells) {
        int gx = cell % G;
        int t = cell / G;
        int gy = t % G;  t /= G;
        int a = t % 3;   int b = t / 3;
        const float* p = &tile[tid * 25];   // stride 25: conflict-free (gcd(25,64)=1)
        float pconf = p[4];
        int m = winner[cell];
        if (m >= 0) {
            const float* g = gts + (size_t)(b * 300 + m) * 25;
            float mx = g[0], my = g[1];
            float w = g[2] * IMG, h = g[3] * IMG;
            float aw = (a == 0) ? aw0 : ((a == 1) ? aw1 : aw2);
            float ah = (a == 0) ? ah0 : ((a == 1) ? ah1 : ah2);
            float tx = mx * (float)G - (float)gx;
            float ty = my * (float)G - (float)gy;
            float tw = logf(w / aw + 1e-16f);
            float th = logf(h / ah + 1e-16f);
            float dx = p[0] - tx, dy = p[1] - ty, dw = p[2] - tw, dh = p[3] - th;
            cAcc = dx * dx + dy * dy + dw * dw + dh * dh;
            oAcc = bce(pconf, 1.0f);               // bce(pconf*obj, obj) at obj cells
            for (int c = 0; c < NCLS; ++c) clAcc += bce(p[5 + c], g[5 + c]);
        } else {
            nAcc = bce(pconf, 0.0f);               // bce(pconf*noObj, 0) at non-obj cells
        }
    }

    s0[tid] = cAcc; s1[tid] = oAcc; s2[tid] = nAcc; s3[tid] = clAcc;
    __syncthreads();
    for (int off = 128; off > 0; off >>= 1) {
        if (tid < off) {
            s0[tid] += s0[tid + off]; s1[tid] += s1[tid + off];
            s2[tid] += s2[tid + off]; s3[tid] += s3[tid + off];
        }
        __syncthreads();
    }
    if (tid == 0) {
        float* o = partials + (size_t)(blockOffset + blockIdx.x) * 4;
        o[0] = s0[0]; o[1] = s1[0]; o[2] = s2[0]; o[3] = s3[0];
    }
}

// Finalize: chain V_WMMA_F32_16X16X4_F32 accumulations, A = ones(16x4),
// B[k][n] = partials[(base + k)][n] for n<4 (other columns zeroed; columns of D
// are independent, so only n=0..3 matter). D[0][n] = component totals.
// numBlocks must be a multiple of 4.
__global__ void yolo_finalize(const float* __restrict__ partials,
                              float* __restrict__ out, int numBlocks) {
    int lane = threadIdx.x;          // 32 threads, EXEC all ones around WMMA
    int n    = lane & 15;            // column (component index when < 4)
    int half = lane >> 4;
    int nc   = n & 3;                // clamped load index, stays in-bounds
    float keep = (n < 4) ? 1.0f : 0.0f;
    v8f c = {};
    v2f a; a.x = 1.0f; a.y = 1.0f;   // A[m][k] = 1 for all m,k
    for (int base = 0; base < numBlocks; base += 4) {
        int blk0 = base + 2 * half;          // VGPR0 holds K = 2*half
        int blk1 = base + 2 * half + 1;      // VGPR1 holds K = 2*half + 1
        v2f bm;
        bm.x = partials[(size_t)blk0 * 4 + nc] * keep;
        bm.y = partials[(size_t)blk1 * 4 + nc] * keep;
        c = __builtin_amdgcn_wmma_f32_16x16x4_f32(
                false, a, false, bm, (short)0, c, false, false);
    }
    __shared__ float s[16];
    if (lane < 16) s[lane] = c[0];   // VGPR0 of C/D: M=0, N=lane (lanes 0..15)
    __syncthreads();
    if (lane == 0) {
        float coord  = s[0] * 5.0f;   // LAMBDA_COORD
        float objL   = s[1] * 1.0f;   // LAMBDA_OBJ
        float noObjL = s[2] * 0.5f;   // LAMBDA_NOOBJ
        float clsL   = s[3] * 1.0f;   // LAMBDA_CLASS
        out[0] = coord + objL + noObjL + clsL;
        out[1] = coord; out[2] = objL; out[3] = noObjL; out[4] = clsL;
    }
}

extern "C" void kernel_launch(void* const* d_in, const int* in_sizes, int n_in,
                              void* d_out, int out_size, void* d_ws, size_t ws_size,
                              hipStream_t stream) {
    (void)in_sizes; (void)n_in; (void)out_size; (void)ws_size;
    const float* p1  = (const float*)d_in[0];   // (64,3,60,60,25)
    const float* p2  = (const float*)d_in[1];   // (64,3,30,30,25)
    const float* p3  = (const float*)d_in[2];   // (64,3,15,15,25)
    const float* gts = (const float*)d_in[3];   // (64,300,25)
    float* out = (float*)d_out;                 // 5 floats: total, comps[4]

    const int B = 64;
    const int nc1 = B * 3 * 60 * 60;            // 691200 cells (multiple of 256)
    const int nc2 = B * 3 * 30 * 30;            // 172800 (multiple of 256)
    const int nc3 = B * 3 * 15 * 15;            //  43200 (tail block)
    const int nb1 = (nc1 + 255) / 256;          // 2700 blocks
    const int nb2 = (nc2 + 255) / 256;          //  675
    const int nb3 = (nc3 + 255) / 256;          //  169
    const int totalBlocks = nb1 + nb2 + nb3;    // 3544 (multiple of 4)
    const int partialInts = totalBlocks * 4;    // 14176 floats

    float* partials = (float*)d_ws;
    int* w1 = (int*)d_ws + partialInts;
    int* w2 = w1 + nc1;
    int* w3 = w2 + nc2;
    const int totalInts = partialInts + nc1 + nc2 + nc3;   // ~3.7 MB of ws

    yolo_init<<<(totalInts + 255) / 256, 256, 0, stream>>>((int*)d_ws, partialInts, totalInts);
    yolo_build<<<(B * 300 + 255) / 256, 256, 0, stream>>>(gts, w1, w2, w3);
    yolo_scale_loss<<<nb1, 256, 0, stream>>>(p1, gts, w1, partials, 60, nc1, 0,
                                             10.f, 16.f, 33.f, 13.f, 30.f, 23.f);
    yolo_scale_loss<<<nb2, 256, 0, stream>>>(p2, gts, w2, partials, 30, nc2, nb1,
                                             30.f, 62.f, 59.f, 61.f, 45.f, 119.f);
    yolo_scale_loss<<<nb3, 256, 0, stream>>>(p3, gts, w3, partials, 15, nc3, nb1 + nb2,
                                             116.f, 156.f, 373.f, 90.f, 198.f, 326.f);
    yolo_finalize<<<1, 32, 0, stream>>>(partials, out, totalBlocks);
}